// GenesetDecoder_5420248728199
// MI455X (gfx1250) — compile-verified
//
#include <hip/hip_runtime.h>
#include <hip/hip_bf16.h>

typedef float v2f __attribute__((ext_vector_type(2)));
typedef float v8f __attribute__((ext_vector_type(8)));

#define B_   64
#define I_   256
#define H_   32
#define O_   5000
#define K_   8192      // I_*H_
#define NT   64        // O-tile per block
#define KSPL 4         // split-K factor (64 genesets per slice)
#define ISL  (I_ / KSPL)
#define EPS  1e-5f

// ---------------------------------------------------------------------------
// Kernel 1: h = relu(x*W1+b1); BatchNorm over batch (biased var); write
// normalized activations as row-major [B_, K_] fp32 panel (the GEMM A matrix).
// One block per geneset i. 256 threads: t = g*32 + h. Thread (g,h) owns batch
// rows b = g, g+8, ..., g+56.
// ---------------------------------------------------------------------------
__global__ __launch_bounds__(256)
void bn_hidden_kernel(const float* __restrict__ x,
                      const float* __restrict__ W1,
                      const float* __restrict__ b1,
                      const float* __restrict__ gamma,
                      const float* __restrict__ beta,
                      float* __restrict__ Hn) {
    const int i = blockIdx.x;
    const int t = threadIdx.x;
    const int h = t & 31;
    const int g = t >> 5;

    const float w  = W1[i * H_ + h];
    const float bb = b1[i * H_ + h];

    float vals[8];
    float s1 = 0.f, s2 = 0.f;
#pragma unroll
    for (int j = 0; j < 8; ++j) {
        const int b = g + j * 8;
        float v = fmaxf(x[b * I_ + i] * w + bb, 0.f);
        vals[j] = v;
        s1 += v;
        s2 += v * v;
    }

    __shared__ float S1[8][32];
    __shared__ float S2[8][32];
    __shared__ float MeanS[32];
    __shared__ float RstdS[32];

    S1[g][h] = s1;
    S2[g][h] = s2;
    __syncthreads();

    if (g == 0) {
        float a = 0.f, q = 0.f;
#pragma unroll
        for (int j = 0; j < 8; ++j) { a += S1[j][h]; q += S2[j][h]; }
        const float mean = a * (1.0f / B_);
        const float var  = q * (1.0f / B_) - mean * mean;   // biased variance
        MeanS[h] = mean;
        RstdS[h] = rsqrtf(var + EPS);
    }
    __syncthreads();

    const float mean = MeanS[h];
    const float rstd = RstdS[h];
    const float ga = gamma[i * H_ + h];
    const float be = beta[i * H_ + h];

#pragma unroll
    for (int j = 0; j < 8; ++j) {
        const int b = g + j * 8;
        Hn[b * K_ + i * H_ + h] = (vals[j] - mean) * rstd * ga + be;
    }
}

// ---------------------------------------------------------------------------
// Kernel 2: bias[o] = sum_i b2[i,o] * adj[i,o]
// ---------------------------------------------------------------------------
__global__ __launch_bounds__(256)
void bias_kernel(const float* __restrict__ b2,
                 const float* __restrict__ adj,
                 float* __restrict__ bias) {
    const int o = blockIdx.x * 256 + threadIdx.x;
    if (o < O_) {
        float acc = 0.f;
        for (int i = 0; i < I_; ++i)
            acc += b2[i * O_ + o] * adj[i * O_ + o];
        bias[o] = acc;
    }
}

// ---------------------------------------------------------------------------
// Kernel 3: split-K WMMA GEMM.
// part[split][64,5000] = Hn[:, split-slice] x (W2*adj)[split-slice, :]
// Grid: (ceil(O_/NT), KSPL). Block: 256 threads = 8 waves.
// Wave w: M-tile = (w&3)*16, N-halves at (w>>2)*32 and +16.
// Register-double-buffered global->LDS pipeline: loads for geneset i+1 are
// issued before the WMMA block for geneset i, so the W2 HBM stream overlaps
// the matrix math.
// ---------------------------------------------------------------------------
__global__ __launch_bounds__(256)
void geneset_gemm_kernel(const float* __restrict__ Hn,
                         const float* __restrict__ W2,
                         const float* __restrict__ adj,
                         float* __restrict__ parts) {
    const int o0 = blockIdx.x * NT;
    const int i0 = blockIdx.y * ISL;
    float* __restrict__ outp = parts + (size_t)blockIdx.y * (B_ * O_);

    const int t    = threadIdx.x;
    const int lane = t & 31;
    const int wave = t >> 5;
    const int m0   = (wave & 3) * 16;
    const int nb   = (wave >> 2) * 32;
    const int l16  = lane & 15;
    const int khi  = (lane >> 4) << 1;   // 0 for lanes 0-15, 2 for lanes 16-31

    __shared__ float As[64][33];    // A tile: [b][k-local], padded
    __shared__ float Bs[32][65];    // B tile: [k-local][o-local], padded

    // Per-thread staging coordinates (2 float4 for A, 2 for B).
    const int flat0 = t * 4;            // 0..1020
    const int flat1 = (t + 256) * 4;    // 1024..2044
    const int brow0 = flat0 >> 5, hc0 = flat0 & 31;
    const int brow1 = flat1 >> 5, hc1 = flat1 & 31;
    const int bh0   = flat0 >> 6, bc0 = flat0 & 63;
    const int bh1   = flat1 >> 6, bc1 = flat1 & 63;

    float4 aR[2], wR[2], mR[2];

    // ---- prologue: load tiles for first geneset of this slice ----
    {
        const int i = i0;
        aR[0] = *reinterpret_cast<const float4*>(&Hn[brow0 * K_ + i * H_ + hc0]);
        aR[1] = *reinterpret_cast<const float4*>(&Hn[brow1 * K_ + i * H_ + hc1]);
#pragma unroll
        for (int p = 0; p < 2; ++p) {
            const int hh = p ? bh1 : bh0;
            const int cc = p ? bc1 : bc0;
            const int o  = o0 + cc;
            if (o + 3 < O_) {
                wR[p] = *reinterpret_cast<const float4*>(&W2[(i * H_ + hh) * O_ + o]);
                mR[p] = *reinterpret_cast<const float4*>(&adj[i * O_ + o]);
            } else {
                float* wp = reinterpret_cast<float*>(&wR[p]);
                float* ap = reinterpret_cast<float*>(&mR[p]);
#pragma unroll
                for (int e = 0; e < 4; ++e) {
                    const int oe = o + e;
                    wp[e] = (oe < O_) ? W2[(i * H_ + hh) * O_ + oe] : 0.f;
                    ap[e] = (oe < O_) ? adj[i * O_ + oe] : 0.f;
                }
            }
        }
    }

    v8f c0 = {};
    v8f c1 = {};

    for (int ii = 0; ii < ISL; ++ii) {
        __syncthreads();   // previous iteration's LDS reads complete

        // ---- publish registers (geneset i0+ii) into LDS ----
        As[brow0][hc0 + 0] = aR[0].x; As[brow0][hc0 + 1] = aR[0].y;
        As[brow0][hc0 + 2] = aR[0].z; As[brow0][hc0 + 3] = aR[0].w;
        As[brow1][hc1 + 0] = aR[1].x; As[brow1][hc1 + 1] = aR[1].y;
        As[brow1][hc1 + 2] = aR[1].z; As[brow1][hc1 + 3] = aR[1].w;

        Bs[bh0][bc0 + 0] = wR[0].x * mR[0].x;
        Bs[bh0][bc0 + 1] = wR[0].y * mR[0].y;
        Bs[bh0][bc0 + 2] = wR[0].z * mR[0].z;
        Bs[bh0][bc0 + 3] = wR[0].w * mR[0].w;
        Bs[bh1][bc1 + 0] = wR[1].x * mR[1].x;
        Bs[bh1][bc1 + 1] = wR[1].y * mR[1].y;
        Bs[bh1][bc1 + 2] = wR[1].z * mR[1].z;
        Bs[bh1][bc1 + 3] = wR[1].w * mR[1].w;

        __syncthreads();

        // ---- issue next-geneset global loads (overlap with WMMA below) ----
        if (ii + 1 < ISL) {
            const int i = i0 + ii + 1;
            aR[0] = *reinterpret_cast<const float4*>(&Hn[brow0 * K_ + i * H_ + hc0]);
            aR[1] = *reinterpret_cast<const float4*>(&Hn[brow1 * K_ + i * H_ + hc1]);
#pragma unroll
            for (int p = 0; p < 2; ++p) {
                const int hh = p ? bh1 : bh0;
                const int cc = p ? bc1 : bc0;
                const int o  = o0 + cc;
                if (o + 3 < O_) {
                    wR[p] = *reinterpret_cast<const float4*>(&W2[(i * H_ + hh) * O_ + o]);
                    mR[p] = *reinterpret_cast<const float4*>(&adj[i * O_ + o]);
                } else {
                    float* wp = reinterpret_cast<float*>(&wR[p]);
                    float* ap = reinterpret_cast<float*>(&mR[p]);
#pragma unroll
                    for (int e = 0; e < 4; ++e) {
                        const int oe = o + e;
                        wp[e] = (oe < O_) ? W2[(i * H_ + hh) * O_ + oe] : 0.f;
                        ap[e] = (oe < O_) ? adj[i * O_ + oe] : 0.f;
                    }
                }
                // deepen the HBM stream: prefetch W2 rows of geneset i+H_ (32 past the one just loaded)
                if (ii + 2 < ISL)
                    __builtin_prefetch(&W2[((i + H_) * H_ + hh) * O_ + o], 0, 3);
            }
        }

        // ---- compute: K-chunk of 32 as 8 steps of 16x16x4, 2 N-subtiles ----
        // A frag (16x4 f32): v[0] = A[M=l16][kk+khi], v[1] = A[..][kk+khi+1]
        // B frag (4x16 f32): v[0] = B[kk+khi][n + l16], v[1] = B[kk+khi+1][..]
#pragma unroll
        for (int kk = 0; kk < H_; kk += 4) {
            v2f a;
            a[0] = As[m0 + l16][kk + khi];
            a[1] = As[m0 + l16][kk + khi + 1];

            v2f bfa;
            bfa[0] = Bs[kk + khi][nb + l16];
            bfa[1] = Bs[kk + khi + 1][nb + l16];
            c0 = __builtin_amdgcn_wmma_f32_16x16x4_f32(
                     false, a, false, bfa, (short)0, c0, false, false);

            v2f bfb;
            bfb[0] = Bs[kk + khi][nb + 16 + l16];
            bfb[1] = Bs[kk + khi + 1][nb + 16 + l16];
            c1 = __builtin_amdgcn_wmma_f32_16x16x4_f32(
                     false, a, false, bfb, (short)0, c1, false, false);
        }
    }

    // ---- store partials: C/D layout VGPR r -> M = m0 + r + (lane>=16 ? 8:0)
#pragma unroll
    for (int r = 0; r < 8; ++r) {
        const int row = m0 + r + ((lane >> 4) << 3);
        const int oo0 = o0 + nb + l16;
        const int oo1 = oo0 + 16;
        if (oo0 < O_) outp[row * O_ + oo0] = c0[r];
        if (oo1 < O_) outp[row * O_ + oo1] = c1[r];
    }
}

// ---------------------------------------------------------------------------
// Kernel 4: out = sum over split-K partials + bias broadcast.
// B_*O_ = 320000 = 1250 * 256 exactly.
// ---------------------------------------------------------------------------
__global__ __launch_bounds__(256)
void reduce_kernel(const float* __restrict__ parts,
                   const float* __restrict__ bias,
                   float* __restrict__ out) {
    const int idx = blockIdx.x * 256 + threadIdx.x;
    if (idx < B_ * O_) {
        const int o = idx % O_;
        float v = bias[o];
#pragma unroll
        for (int s = 0; s < KSPL; ++s)
            v += parts[(size_t)s * (B_ * O_) + idx];
        out[idx] = v;
    }
}

// ---------------------------------------------------------------------------
extern "C" void kernel_launch(void* const* d_in, const int* in_sizes, int n_in,
                              void* d_out, int out_size, void* d_ws, size_t ws_size,
                              hipStream_t stream) {
    const float* x     = (const float*)d_in[0];
    const float* W1    = (const float*)d_in[1];
    const float* b1    = (const float*)d_in[2];
    const float* gamma = (const float*)d_in[3];
    const float* beta  = (const float*)d_in[4];
    const float* W2    = (const float*)d_in[5];
    const float* b2    = (const float*)d_in[6];
    const float* adj   = (const float*)d_in[7];
    float* out = (float*)d_out;

    // workspace layout: Hn [B_,K_] | parts [KSPL][B_,O_] | bias [O_]
    float* Hn    = (float*)d_ws;                       // 2 MB (L2-resident)
    float* parts = Hn + (size_t)B_ * K_;               // 5 MB
    float* bias  = parts + (size_t)KSPL * B_ * O_;     // 20 KB

    bn_hidden_kernel<<<I_, 256, 0, stream>>>(x, W1, b1, gamma, beta, Hn);
    bias_kernel<<<(O_ + 255) / 256, 256, 0, stream>>>(b2, adj, bias);

    dim3 grid((O_ + NT - 1) / NT, KSPL);               // 79 x 4 = 316 blocks
    geneset_gemm_kernel<<<grid, 256, 0, stream>>>(Hn, W2, adj, parts);

    reduce_kernel<<<(B_ * O_) / 256, 256, 0, stream>>>(parts, bias, out);
}